// FruitsClassifierWithSharing_57269093925148
// MI455X (gfx1250) — compile-verified
//
#include <hip/hip_runtime.h>

// Tversky 3-layer net for MI455X (gfx1250).
// With alpha=beta=0.5:  out[b,o] = (Sx[b]+Sw[o]-SA[b,o]) / (Sx[b]+Sw[o]+2eps)
// where SA = sum_k |x_k - w_k|  (L1 distance matrix),
//       Sx/Sw are plain row sums (computed on the WMMA pipe).

typedef __attribute__((ext_vector_type(2))) float v2f;
typedef __attribute__((ext_vector_type(8))) float v8f;

#define TWO_EPS 2e-8f

// ---------------- CDNA5 async global->LDS path (probe-guarded) ----------------
#if defined(__gfx1250__) && __has_builtin(__builtin_amdgcn_global_load_async_to_lds_b32)
#define USE_ASYNC_LDS 1
#else
#define USE_ASYNC_LDS 0
#endif

#if USE_ASYNC_LDS
typedef __attribute__((address_space(1))) int gas_int;  // global
typedef __attribute__((address_space(3))) int las_int;  // LDS
#endif

__device__ __forceinline__ void g2l_b32(const float* g, float* l) {
#if USE_ASYNC_LDS
  // two-step casts: reinterpret pointee (float->int), then addrspacecast
  int* gi = (int*)const_cast<float*>(g);
  int* li = (int*)l;
  __builtin_amdgcn_global_load_async_to_lds_b32((gas_int*)gi, (las_int*)li, 0, 0);
#else
  *l = *g;
#endif
}

__device__ __forceinline__ void wait_g2l() {
#if USE_ASYNC_LDS
#if __has_builtin(__builtin_amdgcn_s_wait_asynccnt)
  __builtin_amdgcn_s_wait_asynccnt(0);
#else
  asm volatile("s_wait_asynccnt 0" ::: "memory");
#endif
#endif
}

// ---------------- Row sums via V_WMMA_F32_16X16X4_F32 ----------------
// One wave32 computes exact f32 row sums for 16 rows: A = 16x4 chunk of M,
// B = ones(4x16)  =>  D[m,n] accumulates rowsum(m) replicated over n.
// A layout (ISA 7.12.2): lanes 0-15 hold K=0,1 (VGPR0,1); lanes 16-31 hold K=2,3.
__global__ __launch_bounds__(256) void rowsum_wmma_kernel(
    const float* __restrict__ M, float* __restrict__ S, int D) {
  const int tid  = threadIdx.x;
  const int lane = tid & 31;
  const int wave = (blockIdx.x * 256 + tid) >> 5;
  const int row_base = wave * 16;           // grid sized so rows = 128 * gridDim
  const int r    = row_base + (lane & 15);
  const int kofs = (lane >> 4) << 1;        // 0 for lanes 0-15, 2 for lanes 16-31

  const v2f b_ones = {1.0f, 1.0f};
  v8f c = {};
  const float* p = M + (size_t)r * D + kofs;
  for (int k = 0; k < D; k += 4) {
    v2f a;
    a.x = p[k];
    a.y = p[k + 1];
    c = __builtin_amdgcn_wmma_f32_16x16x4_f32(
        /*neg_a=*/false, a, /*neg_b=*/false, b_ones,
        /*c_mod=*/(short)0, c, /*reuse_a=*/false, /*reuse_b=*/false);
  }
  // D layout: VGPR i holds M=i (lanes 0-15) and M=i+8 (lanes 16-31); all N equal.
  if (lane == 0) {
#pragma unroll
    for (int i = 0; i < 8; ++i) S[row_base + i] = c[i];
  } else if (lane == 16) {
#pragma unroll
    for (int i = 0; i < 8; ++i) S[row_base + 8 + i] = c[i];
  }
}

// ---------------- L1-distance tile kernel with fused Tversky epilogue ----------------
// grid = (O/BN, B/BM), 256 threads, each thread owns a TM x TN register tile.
// LDS tiles stored transposed [k][row] so compute reads contiguous floats.
template <int BM, int BN, int KC, int TM, int TN, bool RELU>
__global__ __launch_bounds__(256) void tversky_layer_kernel(
    const float* __restrict__ X, const float* __restrict__ W,
    const float* __restrict__ Sx, const float* __restrict__ Sw,
    float* __restrict__ Out, int D, int O) {
  constexpr int THREADS = 256;
  static_assert((BM / TM) * (BN / TN) == THREADS, "thread tiling mismatch");
  __shared__ float Xs[2][KC][BM];
  __shared__ float Ws[2][KC][BN];

  const int tid  = threadIdx.x;
  const int tx   = tid % (BN / TN);
  const int ty   = tid / (BN / TN);
  const int row0 = blockIdx.y * BM;
  const int col0 = blockIdx.x * BN;

  const float* Xg = X + (size_t)row0 * D;
  const float* Wg = W + (size_t)col0 * D;

  constexpr int XE = (BM * KC) / THREADS;
  constexpr int WE = (BN * KC) / THREADS;

  auto stage = [&](int buf, int k0) {
#pragma unroll
    for (int i = 0; i < XE; ++i) {
      int f = tid + i * THREADS;
      int r = f / KC, c = f % KC;                  // coalesced global reads
      g2l_b32(Xg + (size_t)r * D + k0 + c, &Xs[buf][c][r]);
    }
#pragma unroll
    for (int i = 0; i < WE; ++i) {
      int f = tid + i * THREADS;
      int r = f / KC, c = f % KC;
      g2l_b32(Wg + (size_t)r * D + k0 + c, &Ws[buf][c][r]);
    }
  };

  float acc[TM][TN];
#pragma unroll
  for (int i = 0; i < TM; ++i)
#pragma unroll
    for (int j = 0; j < TN; ++j) acc[i][j] = 0.0f;

  stage(0, 0);
  wait_g2l();
  __syncthreads();

  const int NK = D / KC;
  for (int kt = 0; kt < NK; ++kt) {
    const int cur = kt & 1;
    if (kt + 1 < NK) stage(cur ^ 1, (kt + 1) * KC);  // prefetch next tile (async)
#pragma unroll
    for (int k = 0; k < KC; ++k) {
      float xv[TM], wv[TN];
#pragma unroll
      for (int i = 0; i < TM; ++i) xv[i] = Xs[cur][k][ty * TM + i];
#pragma unroll
      for (int j = 0; j < TN; ++j) wv[j] = Ws[cur][k][tx * TN + j];
#pragma unroll
      for (int i = 0; i < TM; ++i)
#pragma unroll
        for (int j = 0; j < TN; ++j)
          acc[i][j] += __builtin_fabsf(xv[i] - wv[j]);  // v_sub + v_add |src|
    }
    if (kt + 1 < NK) {
      wait_g2l();
      __syncthreads();
    }
  }

  // Fused Tversky ratio + optional ReLU epilogue.
  float sxr[TM], swc[TN];
#pragma unroll
  for (int i = 0; i < TM; ++i) sxr[i] = Sx[row0 + ty * TM + i];
#pragma unroll
  for (int j = 0; j < TN; ++j) swc[j] = Sw[col0 + tx * TN + j];
#pragma unroll
  for (int i = 0; i < TM; ++i) {
#pragma unroll
    for (int j = 0; j < TN; ++j) {
      float s = sxr[i] + swc[j];
      float v = (s - acc[i][j]) / (s + TWO_EPS);
      if (RELU) v = fmaxf(v, 0.0f);
      Out[(size_t)(row0 + ty * TM + i) * O + (col0 + tx * TN + j)] = v;
    }
  }
}

// ---------------- Host orchestration ----------------
extern "C" void kernel_launch(void* const* d_in, const int* in_sizes, int n_in,
                              void* d_out, int out_size, void* d_ws, size_t ws_size,
                              hipStream_t stream) {
  (void)in_sizes; (void)n_in; (void)out_size; (void)ws_size;
  const float* x  = (const float*)d_in[0];   // [256,1024]
  const float* W1 = (const float*)d_in[1];   // [1024,1024]
  const float* W2 = (const float*)d_in[2];   // [1024,1024]
  const float* W3 = (const float*)d_in[3];   // [128,1024]
  float* out = (float*)d_out;                // [256,128]

  constexpr int B = 256, D = 1024, H = 1024, C = 128;

  float* ws  = (float*)d_ws;
  float* H1  = ws;            // 256*1024
  float* H2  = H1 + B * H;    // 256*1024
  float* Sx  = H2 + B * H;    // 256
  float* Sw1 = Sx + B;        // 1024
  float* Sw2 = Sw1 + H;       // 1024
  float* Sw3 = Sw2 + H;       // 128
  float* Sh1 = Sw3 + C;       // 256
  float* Sh2 = Sh1 + B;       // 256

  // Row sums on the WMMA pipe (128 rows per 256-thread block).
  rowsum_wmma_kernel<<<B / 128, 256, 0, stream>>>(x,  Sx,  D);
  rowsum_wmma_kernel<<<H / 128, 256, 0, stream>>>(W1, Sw1, D);
  rowsum_wmma_kernel<<<H / 128, 256, 0, stream>>>(W2, Sw2, H);
  rowsum_wmma_kernel<<<C / 128, 256, 0, stream>>>(W3, Sw3, H);

  // Layer 1: [256,1024] x [1024,1024] -> relu
  tversky_layer_kernel<64, 64, 16, 4, 4, true>
      <<<dim3(H / 64, B / 64), 256, 0, stream>>>(x, W1, Sx, Sw1, H1, D, H);
  rowsum_wmma_kernel<<<B / 128, 256, 0, stream>>>(H1, Sh1, H);

  // Layer 2: [256,1024] x [1024,1024] -> relu
  tversky_layer_kernel<64, 64, 16, 4, 4, true>
      <<<dim3(H / 64, B / 64), 256, 0, stream>>>(H1, W2, Sh1, Sw2, H2, H, H);
  rowsum_wmma_kernel<<<B / 128, 256, 0, stream>>>(H2, Sh2, H);

  // Layer 3: [256,1024] x [128,1024] -> no relu (smaller tiles for parallelism)
  tversky_layer_kernel<32, 32, 16, 2, 2, false>
      <<<dim3(C / 32, B / 32), 256, 0, stream>>>(H2, W3, Sh2, Sw3, out, H, C);
}